// PixelContrastLoss_36936718745849
// MI455X (gfx1250) — compile-verified
//
#include <hip/hip_runtime.h>
#include <math.h>

typedef __attribute__((ext_vector_type(2))) float v2f;
typedef __attribute__((ext_vector_type(4))) float v4f;
typedef __attribute__((ext_vector_type(8))) float v8f;

#define TOTAL_CLASSES 64
#define N_VIEW        128
#define FEAT_DIM      128
#define N_ROWS        (TOTAL_CLASSES * N_VIEW)   // 8192
#define CHUNK_COLS    64
#define LDS_STRIDE    (FEAT_DIM + 4)             // 132 dwords: 16B-aligned rows, 4-bank lane stride
#define SPLITS        8
#define COLS_PER_SPLIT (N_ROWS / SPLITS)         // 1024
#define CHUNKS_PER_SPLIT (COLS_PER_SPLIT / CHUNK_COLS) // 16
#define TILES_PER_CHUNK (CHUNK_COLS / 16)        // 4
#define NEG_INF (-3.402823466e38f)

// ---------------------------------------------------------------------------
// Kernel 1: L2-normalize each of the 8192 rows (one wave32 per row) and
// initialize rowmax[] to -inf for the atomic-max pass.
// ---------------------------------------------------------------------------
__global__ __launch_bounds__(256) void pcl_normalize(const float* __restrict__ x,
                                                     float* __restrict__ featn,
                                                     float* __restrict__ rowmax) {
    const int lane = threadIdx.x & 31;
    const int wave = threadIdx.x >> 5;
    const int row  = blockIdx.x * 8 + wave;

    v4f v = *(const v4f*)(x + (size_t)row * FEAT_DIM + lane * 4);
    float ss = v.x * v.x + v.y * v.y + v.z * v.z + v.w * v.w;
    ss += __shfl_xor(ss, 16, 32);
    ss += __shfl_xor(ss, 8, 32);
    ss += __shfl_xor(ss, 4, 32);
    ss += __shfl_xor(ss, 2, 32);
    ss += __shfl_xor(ss, 1, 32);
    const float inv = 1.0f / fmaxf(sqrtf(ss), 1e-12f);
    v4f o;
    o.x = v.x * inv; o.y = v.y * inv; o.z = v.z * inv; o.w = v.w * inv;
    *(v4f*)(featn + (size_t)row * FEAT_DIM + lane * 4) = o;
    if (lane == 0) rowmax[row] = NEG_INF;
}

// ---------------------------------------------------------------------------
// Shared GEMM machinery: one wave owns a 16-row tile (full K=128 A held in
// 64 VGPRs), B tiles come from an LDS-staged 64-column chunk that is filled
// by CDNA5 async global->LDS copies (ASYNCcnt-tracked, no VGPR staging).
// Gram symmetry: A and B per-lane WMMA layouts are identical:
//   vgpr0 = F[base + (lane&15)][4k + ((lane&16)?2:0)], vgpr1 = +1
// ---------------------------------------------------------------------------
__device__ __forceinline__ void load_a_tile(const float* __restrict__ featn,
                                            int rowBase, int lane, int koff2,
                                            v2f* __restrict__ Areg) {
    const float* arow =
        featn + (size_t)(rowBase + (lane & 15)) * FEAT_DIM + koff2;
#pragma unroll
    for (int kk = 0; kk < 32; ++kk) Areg[kk] = *(const v2f*)(arow + 4 * kk);
}

// Issue 8 async B128 copies per thread: 64 cols x 128 floats -> padded LDS.
// ldsBase is the LDS byte address of Bs[0] (low 32 bits of the flat pointer).
__device__ __forceinline__ void load_chunk_async(const float* __restrict__ featn,
                                                 unsigned ldsBase,
                                                 int colChunkBase, int tid) {
#pragma unroll
    for (int i = 0; i < 8; ++i) {
        int t  = tid + i * 256;
        int c  = t >> 5;            // 0..63 column within chunk
        int kq = (t & 31) << 2;     // 0..124 step 4
        unsigned lds = ldsBase + (unsigned)(c * LDS_STRIDE + kq) * 4u;
        const float* g =
            featn + (size_t)(colChunkBase + c) * FEAT_DIM + kq;
        asm volatile("global_load_async_to_lds_b128 %0, %1, off"
                     :
                     : "v"(lds), "v"((unsigned long long)(uintptr_t)g)
                     : "memory");
    }
}

__device__ __forceinline__ void wait_async_and_barrier() {
    asm volatile("s_wait_asynccnt 0x0" ::: "memory");
    __syncthreads();
}

__device__ __forceinline__ v8f sim_tile(const v2f* __restrict__ Areg,
                                        const float* __restrict__ Bs,
                                        int bbase) {
    v8f c = {};
#pragma unroll
    for (int kk = 0; kk < 32; ++kk) {
        v2f b = *(const v2f*)(Bs + bbase + 4 * kk);
        c = __builtin_amdgcn_wmma_f32_16x16x4_f32(
            /*neg_a=*/false, Areg[kk], /*neg_b=*/false, b,
            /*c_mod=*/(short)0, c, /*reuse_a=*/false, /*reuse_b=*/false);
    }
    return c;
}

// ---------------------------------------------------------------------------
// Kernel 2: partial row-max over a 1024-column split; combined with an exact
// (deterministic) float atomic max.  grid = (64 row-tile blocks, 8 splits).
// ---------------------------------------------------------------------------
__global__ __launch_bounds__(256) void pcl_rowmax(const float* __restrict__ featn,
                                                  float* __restrict__ rowmax) {
    __shared__ float Bs[CHUNK_COLS * LDS_STRIDE];
    const int tid     = threadIdx.x;
    const int lane    = tid & 31;
    const int wave    = tid >> 5;
    const int rowBase = (blockIdx.x * 8 + wave) * 16;
    const int split   = blockIdx.y;
    const int koff2   = (lane & 16) ? 2 : 0;
    const unsigned ldsBase = (unsigned)(uintptr_t)(void*)Bs;

    v2f Areg[32];
    load_a_tile(featn, rowBase, lane, koff2, Areg);

    float vmax[8];
#pragma unroll
    for (int r = 0; r < 8; ++r) vmax[r] = NEG_INF;

    for (int ch = 0; ch < CHUNKS_PER_SPLIT; ++ch) {
        const int colChunkBase = split * COLS_PER_SPLIT + ch * CHUNK_COLS;
        __syncthreads();                       // previous chunk fully consumed
        load_chunk_async(featn, ldsBase, colChunkBase, tid);
        if (ch + 1 < CHUNKS_PER_SPLIT)
            __builtin_prefetch(featn + (size_t)(colChunkBase + CHUNK_COLS) * FEAT_DIM, 0, 0);
        wait_async_and_barrier();              // all async writes visible
#pragma unroll
        for (int ct = 0; ct < TILES_PER_CHUNK; ++ct) {
            const int bbase = (ct * 16 + (lane & 15)) * LDS_STRIDE + koff2;
            v8f c = sim_tile(Areg, Bs, bbase);
#pragma unroll
            for (int r = 0; r < 8; ++r) vmax[r] = fmaxf(vmax[r], c[r]);
        }
    }

#pragma unroll
    for (int r = 0; r < 8; ++r) {
        float v = vmax[r];
        v = fmaxf(v, __shfl_xor(v, 1, 32));
        v = fmaxf(v, __shfl_xor(v, 2, 32));
        v = fmaxf(v, __shfl_xor(v, 4, 32));
        v = fmaxf(v, __shfl_xor(v, 8, 32));
        if ((lane & 15) == 0) {
            const int rowIdx = rowBase + r + ((lane >> 4) << 3);
            __hip_atomic_fetch_max(&rowmax[rowIdx], v, __ATOMIC_RELAXED,
                                   __HIP_MEMORY_SCOPE_AGENT);
        }
    }
}

// ---------------------------------------------------------------------------
// Kernel 3: recompute sim tiles, accumulate masked exp-sums per split into
// private partial buffers (no float-add atomics -> deterministic).
// mask[i,j] = (y[i>>7] == y[j & 63]); col label pattern repeats every 64 cols.
// ---------------------------------------------------------------------------
__global__ __launch_bounds__(256) void pcl_losspart(const float* __restrict__ featn,
                                                    const float* __restrict__ rowmax,
                                                    const int* __restrict__ y,
                                                    float* __restrict__ posp,
                                                    float* __restrict__ negp) {
    __shared__ float Bs[CHUNK_COLS * LDS_STRIDE];
    const int tid     = threadIdx.x;
    const int lane    = tid & 31;
    const int wave    = tid >> 5;
    const int rowBase = (blockIdx.x * 8 + wave) * 16;
    const int split   = blockIdx.y;
    const int koff2   = (lane & 16) ? 2 : 0;
    const unsigned ldsBase = (unsigned)(uintptr_t)(void*)Bs;

    v2f Areg[32];
    load_a_tile(featn, rowBase, lane, koff2, Areg);

    const int rowlab = y[rowBase >> 7];  // uniform across the 16-row tile
    float rm[8];
#pragma unroll
    for (int r = 0; r < 8; ++r)
        rm[r] = rowmax[rowBase + r + ((lane >> 4) << 3)];

    int cl[TILES_PER_CHUNK];
#pragma unroll
    for (int ct = 0; ct < TILES_PER_CHUNK; ++ct)
        cl[ct] = y[ct * 16 + (lane & 15)];  // (colBase + n) & 63 == ct*16 + n

    float pos[8], neg[8];
#pragma unroll
    for (int r = 0; r < 8; ++r) { pos[r] = 0.0f; neg[r] = 0.0f; }

    for (int ch = 0; ch < CHUNKS_PER_SPLIT; ++ch) {
        const int colChunkBase = split * COLS_PER_SPLIT + ch * CHUNK_COLS;
        __syncthreads();                       // previous chunk fully consumed
        load_chunk_async(featn, ldsBase, colChunkBase, tid);
        if (ch + 1 < CHUNKS_PER_SPLIT)
            __builtin_prefetch(featn + (size_t)(colChunkBase + CHUNK_COLS) * FEAT_DIM, 0, 0);
        wait_async_and_barrier();              // all async writes visible
#pragma unroll
        for (int ct = 0; ct < TILES_PER_CHUNK; ++ct) {
            const int bbase = (ct * 16 + (lane & 15)) * LDS_STRIDE + koff2;
            v8f c = sim_tile(Areg, Bs, bbase);
            const bool ispos = (cl[ct] == rowlab);
#pragma unroll
            for (int r = 0; r < 8; ++r) {
                const float e = __expf(c[r] - rm[r]);
                pos[r] += ispos ? e : 0.0f;
                neg[r] += ispos ? 0.0f : e;
            }
        }
    }

#pragma unroll
    for (int r = 0; r < 8; ++r) {
        float p = pos[r], q = neg[r];
        p += __shfl_xor(p, 1, 32); q += __shfl_xor(q, 1, 32);
        p += __shfl_xor(p, 2, 32); q += __shfl_xor(q, 2, 32);
        p += __shfl_xor(p, 4, 32); q += __shfl_xor(q, 4, 32);
        p += __shfl_xor(p, 8, 32); q += __shfl_xor(q, 8, 32);
        if ((lane & 15) == 0) {
            const int rowIdx = rowBase + r + ((lane >> 4) << 3);
            posp[(size_t)split * N_ROWS + rowIdx] = p;
            negp[(size_t)split * N_ROWS + rowIdx] = q;
        }
    }
}

// ---------------------------------------------------------------------------
// Kernel 4: ordered combine of the 8 partials per row -> per-row loss.
// ---------------------------------------------------------------------------
__global__ __launch_bounds__(256) void pcl_lossfinal(const float* __restrict__ posp,
                                                     const float* __restrict__ negp,
                                                     float* __restrict__ lossv) {
    const int row = blockIdx.x * 256 + threadIdx.x;
    float p = 0.0f, q = 0.0f;
#pragma unroll
    for (int s = 0; s < SPLITS; ++s) {
        p += posp[(size_t)s * N_ROWS + row];
        q += negp[(size_t)s * N_ROWS + row];
    }
    lossv[row] = -logf(p / (p + q + 1e-8f) + 1e-8f);
}

// ---------------------------------------------------------------------------
// Kernel 5: deterministic mean over 8192 losses.
// ---------------------------------------------------------------------------
__global__ __launch_bounds__(256) void pcl_mean(const float* __restrict__ lossv,
                                                float* __restrict__ out) {
    __shared__ float red[256];
    float s = 0.0f;
    for (int i = threadIdx.x; i < N_ROWS; i += 256) s += lossv[i];
    red[threadIdx.x] = s;
    for (int w = 128; w > 0; w >>= 1) {
        __syncthreads();
        if (threadIdx.x < w) red[threadIdx.x] += red[threadIdx.x + w];
    }
    if (threadIdx.x == 0) out[0] = red[0] * (1.0f / (float)N_ROWS);
}

// ---------------------------------------------------------------------------
extern "C" void kernel_launch(void* const* d_in, const int* in_sizes, int n_in,
                              void* d_out, int out_size, void* d_ws, size_t ws_size,
                              hipStream_t stream) {
    (void)in_sizes; (void)n_in; (void)out_size; (void)ws_size;
    const float* x = (const float*)d_in[0];
    const int*   y = (const int*)d_in[1];

    float* featn  = (float*)d_ws;                               // 8192*128 f32 (4 MB)
    float* rowmax = featn + (size_t)N_ROWS * FEAT_DIM;          // 8192 f32
    float* posp   = rowmax + N_ROWS;                            // 8*8192 f32
    float* negp   = posp + (size_t)SPLITS * N_ROWS;             // 8*8192 f32
    float* lossv  = negp + (size_t)SPLITS * N_ROWS;             // 8192 f32

    dim3 gGemm(N_ROWS / (16 * 8), SPLITS);  // (64, 8) blocks of 8 waves

    pcl_normalize<<<N_ROWS / 8, 256, 0, stream>>>(x, featn, rowmax);
    pcl_rowmax<<<gGemm, 256, 0, stream>>>(featn, rowmax);
    pcl_losspart<<<gGemm, 256, 0, stream>>>(featn, rowmax, y, posp, negp);
    pcl_lossfinal<<<N_ROWS / 256, 256, 0, stream>>>(posp, negp, lossv);
    pcl_mean<<<1, 256, 0, stream>>>(lossv, (float*)d_out);
}